// GraphUnet_47132971106799
// MI455X (gfx1250) — compile-verified
//
#include <hip/hip_runtime.h>
#include <math.h>

#define GAT_ALPHA 0.2f

typedef float v2f __attribute__((ext_vector_type(2)));
typedef float v8f __attribute__((ext_vector_type(8)));

// ======================= WMMA f32 GEMM (N fixed = 64) =====================
// C(Mx64) = A(MxK) @ B(Kx64), row-major, M%64==0, K%4==0.
// One wave32 computes a 16x64 strip: one A fragment (16x4) is reused by
// 4 V_WMMA_F32_16X16X4_F32 ops per K-step (4 accumulators) so the big A
// operand (att matrix, up to 67MB) is streamed from HBM exactly once.
// act: 0 = none, 1 = ELU.
__global__ void wmma_gemm_f32(const float* __restrict__ A,
                              const float* __restrict__ B,
                              float* __restrict__ C,
                              int M, int K, int act) {
  const int lane = threadIdx.x;   // 0..31 (wave32)
  const int half = lane >> 4;     // 0: lanes 0-15 (K=0,1), 1: lanes 16-31 (K=2,3)
  const int l    = lane & 15;
  const int m0 = (blockIdx.x * blockDim.y + threadIdx.y) * 16;
  const float* Arow = A + (long)(m0 + l) * K + 2 * half;   // A 16x4 lane layout
  const float* Brow = B + (long)(2 * half) * 64 + l;       // B 4x16 lane layout
  v8f acc0 = {}, acc1 = {}, acc2 = {}, acc3 = {};
  for (int k = 0; k < K; k += 4) {
    if ((k & 28) == 0)                      // once per 128B of per-lane A stream
      __builtin_prefetch(Arow + k + 512, 0, 1);
    v2f a;
    a.x = Arow[k];
    a.y = Arow[k + 1];
    const float* b0 = Brow + (long)k * 64;
    const float* b1 = b0 + 64;
    v2f bt;
    bt.x = b0[0];  bt.y = b1[0];
    acc0 = __builtin_amdgcn_wmma_f32_16x16x4_f32(false, a, false, bt, (short)0, acc0,
                                                 false, false);
    bt.x = b0[16]; bt.y = b1[16];
    acc1 = __builtin_amdgcn_wmma_f32_16x16x4_f32(false, a, false, bt, (short)0, acc1,
                                                 false, false);
    bt.x = b0[32]; bt.y = b1[32];
    acc2 = __builtin_amdgcn_wmma_f32_16x16x4_f32(false, a, false, bt, (short)0, acc2,
                                                 false, false);
    bt.x = b0[48]; bt.y = b1[48];
    acc3 = __builtin_amdgcn_wmma_f32_16x16x4_f32(false, a, false, bt, (short)0, acc3,
                                                 false, false);
  }
  const int row_base = m0 + half * 8;  // C/D layout: VGPR v -> rows m0+v / m0+v+8
#pragma unroll
  for (int v = 0; v < 8; ++v) {
    float r0 = acc0[v], r1 = acc1[v], r2 = acc2[v], r3 = acc3[v];
    if (act == 1) {
      r0 = (r0 > 0.f) ? r0 : (expf(r0) - 1.f);  // ELU
      r1 = (r1 > 0.f) ? r1 : (expf(r1) - 1.f);
      r2 = (r2 > 0.f) ? r2 : (expf(r2) - 1.f);
      r3 = (r3 > 0.f) ? r3 : (expf(r3) - 1.f);
    }
    float* crow = C + (long)(row_base + v) * 64 + l;
    crow[0]  = r0;
    crow[16] = r1;
    crow[32] = r2;
    crow[48] = r3;
  }
}

// ======================= elementwise / utility ============================
__global__ void zero4_kernel(float4* __restrict__ p, long n4) {
  long i = (long)blockIdx.x * blockDim.x + threadIdx.x;
  long stride = (long)gridDim.x * blockDim.x;
  float4 z = {0.f, 0.f, 0.f, 0.f};
  for (; i < n4; i += stride) p[i] = z;
}

__global__ void copy4_kernel(float4* __restrict__ dst, const float4* __restrict__ src,
                             long n4) {
  long i = (long)blockIdx.x * blockDim.x + threadIdx.x;
  long stride = (long)gridDim.x * blockDim.x;
  for (; i < n4; i += stride) dst[i] = src[i];
}

// f1 = h @ a1, f2 = h @ a2  (h: Nx64)
__global__ void attvec_kernel(const float* __restrict__ h, const float* __restrict__ a1,
                              const float* __restrict__ a2, float* __restrict__ f1,
                              float* __restrict__ f2, int N) {
  int i = blockIdx.x * blockDim.x + threadIdx.x;
  if (i >= N) return;
  const float* hr = h + (long)i * 64;
  float s1 = 0.f, s2 = 0.f;
#pragma unroll
  for (int c = 0; c < 64; ++c) {
    float v = hr[c];
    s1 += v * a1[c];
    s2 += v * a2[c];
  }
  f1[i] = s1;
  f2[i] = s2;
}

// scores = sigmoid(x @ p)
__global__ void pool_scores_kernel(const float* __restrict__ x,
                                   const float* __restrict__ p,
                                   float* __restrict__ s, int N) {
  int i = blockIdx.x * blockDim.x + threadIdx.x;
  if (i >= N) return;
  const float* xr = x + (long)i * 64;
  float d = 0.f;
#pragma unroll
  for (int c = 0; c < 64; ++c) d += xr[c] * p[c];
  s[i] = 1.f / (1.f + expf(-d));
}

// ======================= masked row softmax ===============================
// att[i,:] = softmax over j of leakyrelu(f1_i + f2_j) where adj[i,j] > 0.
// Single streaming read of adj; per-thread e-values held in 16 registers
// (N is always a multiple of 256 and <= 4096, so bounds tests are uniform);
// att written exactly once. Masked entries get exact 0 (matches exp(-1e9-max)
// underflow in the reference).
__global__ void att_softmax_kernel(const float* __restrict__ adj,
                                   const float* __restrict__ f1,
                                   const float* __restrict__ f2,
                                   float* __restrict__ att, int N) {
  __shared__ float sdata[256];
  const int row = blockIdx.x;
  const int tid = threadIdx.x;
  const float* arow = adj + (long)row * N;
  float* orow = att + (long)row * N;
  const float f1i = f1[row];
  float ev[16];
  // pass 1 (global read): e-values + row max
  float lmax = -1e30f;
#pragma unroll
  for (int it = 0; it < 16; ++it) {
    int j = it * 256 + tid;
    float e = -1e30f;
    if (j < N && arow[j] > 0.f) {           // j<N uniform per iteration
      e = f1i + f2[j];
      e = (e > 0.f) ? e : GAT_ALPHA * e;
      lmax = fmaxf(lmax, e);
    }
    ev[it] = e;
  }
  sdata[tid] = lmax;
  __syncthreads();
  for (int s = 128; s > 0; s >>= 1) {
    if (tid < s) sdata[tid] = fmaxf(sdata[tid], sdata[tid + s]);
    __syncthreads();
  }
  const float rmax = sdata[0];
  __syncthreads();
  // pass 2 (registers only): exp + sum
  float lsum = 0.f;
#pragma unroll
  for (int it = 0; it < 16; ++it) {
    float w = (ev[it] > -1e29f) ? expf(ev[it] - rmax) : 0.f;
    ev[it] = w;
    lsum += w;
  }
  sdata[tid] = lsum;
  __syncthreads();
  for (int s = 128; s > 0; s >>= 1) {
    if (tid < s) sdata[tid] += sdata[tid + s];
    __syncthreads();
  }
  const float inv = 1.f / sdata[0];
  __syncthreads();
  // pass 3 (single global write)
#pragma unroll
  for (int it = 0; it < 16; ++it) {
    int j = it * 256 + tid;
    if (j < N) orow[j] = ev[it] * inv;
  }
}

// ======================= top-k (bitonic, deterministic) ===================
// Full descending sort of (score, index) with ties broken by lower index,
// matching jax.lax.top_k ordering. N is a power of two <= 4096.
__global__ void topk_kernel(const float* __restrict__ scores, float* __restrict__ vals,
                            int* __restrict__ sidx, int N) {
  __shared__ float s[4096];
  __shared__ int id[4096];
  const int tid = threadIdx.x;
  const int T = blockDim.x;
  for (int i = tid; i < N; i += T) {
    s[i] = scores[i];
    id[i] = i;
  }
  __syncthreads();
  for (int k = 2; k <= N; k <<= 1) {
    for (int j = k >> 1; j > 0; j >>= 1) {
      for (int i = tid; i < N; i += T) {
        int ixj = i ^ j;
        if (ixj > i) {
          float sa = s[i], sb = s[ixj];
          int ia = id[i], ib = id[ixj];
          bool a_before_b = (sa > sb) || (sa == sb && ia < ib);
          bool up = ((i & k) == 0);
          if (up ? !a_before_b : a_before_b) {
            s[i] = sb; s[ixj] = sa;
            id[i] = ib; id[ixj] = ia;
          }
        }
      }
      __syncthreads();
    }
  }
  const int khalf = N >> 1;
  for (int i = tid; i < khalf; i += T) {
    vals[i] = s[i];
    sidx[i] = id[i];
  }
}

// ======================= gathers / scatters ===============================
__global__ void gather_newx_kernel(const float* __restrict__ x, const float* __restrict__ vals,
                                   const int* __restrict__ idx, float* __restrict__ nx, int k) {
  int g = blockIdx.x * blockDim.x + threadIdx.x;
  if (g >= k * 64) return;
  int r = g >> 6, c = g & 63;
  nx[g] = x[(long)idx[r] * 64 + c] * vals[r];
}

__global__ void gather_adj_kernel(const float* __restrict__ adjo, const int* __restrict__ idx,
                                  float* __restrict__ adjn, int k, int No) {
  long g = (long)blockIdx.x * blockDim.x + threadIdx.x;
  if (g >= (long)k * k) return;
  int r = (int)(g / k), c = (int)(g % k);
  adjn[g] = adjo[(long)idx[r] * No + idx[c]];
}

__global__ void compose_idx_kernel(const int* __restrict__ comp_prev,
                                   const int* __restrict__ sidx,
                                   int* __restrict__ comp, int* __restrict__ idx_store,
                                   int k) {
  int r = blockIdx.x * blockDim.x + threadIdx.x;
  if (r >= k) return;
  int sv = sidx[r];
  idx_store[r] = sv;
  comp[r] = comp_prev ? comp_prev[sv] : sv;
}

__global__ void scatter_rows_kernel(const float* __restrict__ src, const int* __restrict__ comp,
                                    float* __restrict__ dst, int k, int C) {
  int g = blockIdx.x * blockDim.x + threadIdx.x;
  if (g >= k * C) return;
  int r = g / C, c = g % C;
  dst[(long)comp[r] * C + c] = src[g];
}

__global__ void scatter_add_rows_kernel(const float* __restrict__ src,
                                        const int* __restrict__ idx,
                                        float* __restrict__ dst, int k, int C) {
  int g = blockIdx.x * blockDim.x + threadIdx.x;
  if (g >= k * C) return;
  int r = g / C, c = g % C;
  dst[(long)idx[r] * C + c] += src[g];  // idx rows are unique -> race free
}

__global__ void scatter_scalar_kernel(const float* __restrict__ src,
                                      const int* __restrict__ comp,
                                      float* __restrict__ dst, int k) {
  int r = blockIdx.x * blockDim.x + threadIdx.x;
  if (r >= k) return;
  dst[comp[r]] = src[r];
}

__global__ void scatter_adj_kernel(const float* __restrict__ adjn, const int* __restrict__ comp,
                                   float* __restrict__ recon, int k, int No) {
  long g = (long)blockIdx.x * blockDim.x + threadIdx.x;
  if (g >= (long)k * k) return;
  int r = (int)(g / k), c = (int)(g % k);
  recon[(long)comp[r] * No + comp[c]] = adjn[g];
}

// out = sigmoid(x @ W_out + b_out), x: Nx64, W: 64x8
__global__ void out_layer_kernel(const float* __restrict__ x, const float* __restrict__ W,
                                 const float* __restrict__ b, float* __restrict__ out, int N) {
  int g = blockIdx.x * blockDim.x + threadIdx.x;
  if (g >= N * 8) return;
  int i = g >> 3, o = g & 7;
  float s = b[o];
  const float* xr = x + (long)i * 64;
#pragma unroll
  for (int c = 0; c < 64; ++c) s += xr[c] * W[c * 8 + o];
  out[g] = 1.f / (1.f + expf(-s));
}

// ======================= host orchestration ===============================
extern "C" void kernel_launch(void* const* d_in, const int* in_sizes, int n_in,
                              void* d_out, int out_size, void* d_ws, size_t ws_size,
                              hipStream_t stream) {
  (void)in_sizes; (void)n_in; (void)out_size; (void)ws_size;
  const float* x_in     = (const float*)d_in[0];
  const float* adj_in   = (const float*)d_in[1];
  const float* W_start  = (const float*)d_in[2];
  const float* a_start  = (const float*)d_in[3];
  const float* W_down   = (const float*)d_in[4];
  const float* a_down   = (const float*)d_in[5];
  const float* W_bottom = (const float*)d_in[6];
  const float* a_bottom = (const float*)d_in[7];
  const float* W_up     = (const float*)d_in[8];
  const float* a_up     = (const float*)d_in[9];
  const float* p_pool   = (const float*)d_in[10];
  const float* W_out    = (const float*)d_in[11];
  const float* b_out    = (const float*)d_in[12];

  const int N0 = 4096, H = 64, F0 = 128;
  float* out = (float*)d_out;
  // output regions (concatenated, reference return order)
  float* out_pred     = out;                                   // 4096*8
  float* out_gft_down = out + 32768;                           // 3 * 4096*64
  float* out_gft_up   = out + 32768 + 3 * 262144;              // 3 * 4096*64
  float* out_att      = out + 32768 + 6 * 262144;              // 4096*4096
  float* out_scores   = out_att + (long)N0 * N0;               // 3 * 4096
  float* out_recon    = out_scores + 3 * N0;                   // 3 * 4096*4096

  // workspace layout (256B aligned slabs), ~95 MB total
  char* wp = (char*)d_ws;
  auto alloc = [&](size_t bytes) -> char* {
    char* p = wp;
    wp += (bytes + 255) & ~(size_t)255;
    return p;
  };
  float* att_ws = (float*)alloc((size_t)N0 * N0 * 4);
  float* adj1   = (float*)alloc((size_t)2048 * 2048 * 4);
  float* adj2   = (float*)alloc((size_t)1024 * 1024 * 4);
  float* adj3   = (float*)alloc((size_t)512 * 512 * 4);
  float* hA     = (float*)alloc((size_t)N0 * H * 4);
  float* hB     = (float*)alloc((size_t)N0 * H * 4);
  float* hC     = (float*)alloc((size_t)N0 * H * 4);
  float* down0  = (float*)alloc((size_t)4096 * H * 4);
  float* down1  = (float*)alloc((size_t)2048 * H * 4);
  float* down2  = (float*)alloc((size_t)1024 * H * 4);
  float* f1b    = (float*)alloc((size_t)N0 * 4);
  float* f2b    = (float*)alloc((size_t)N0 * 4);
  float* scoresb= (float*)alloc((size_t)N0 * 4);
  float* svals  = (float*)alloc((size_t)2048 * 4);
  int*   sidx   = (int*)alloc((size_t)2048 * 4);
  int*   idx0   = (int*)alloc((size_t)2048 * 4);
  int*   idx1   = (int*)alloc((size_t)1024 * 4);
  int*   idx2   = (int*)alloc((size_t)512 * 4);
  int*   comp0  = (int*)alloc((size_t)2048 * 4);
  int*   comp1  = (int*)alloc((size_t)1024 * 4);
  int*   comp2  = (int*)alloc((size_t)512 * 4);

  float* down_arr[3] = {down0, down1, down2};
  float* adj_next_arr[3] = {adj1, adj2, adj3};
  int* idx_arr[3] = {idx0, idx1, idx2};
  int* comp_arr[3] = {comp0, comp1, comp2};

  auto zero = [&](float* p, long n) {  // n % 4 == 0 for all regions used
    hipLaunchKernelGGL(zero4_kernel, dim3(2048), dim3(256), 0, stream, (float4*)p, n / 4);
  };
  auto copy = [&](float* dst, const float* src, long n) {
    hipLaunchKernelGGL(copy4_kernel, dim3(512), dim3(256), 0, stream,
                       (float4*)dst, (const float4*)src, n / 4);
  };
  auto gemm = [&](const float* A, const float* B, float* C, int M, int K, int act) {
    dim3 blk(32, 4);
    dim3 grd((unsigned)(M / 64));
    hipLaunchKernelGGL(wmma_gemm_f32, grd, blk, 0, stream, A, B, C, M, K, act);
  };
  auto gat = [&](const float* x, const float* adj, const float* W, const float* avec,
                 int N, int F, float* attbuf, float* xout) {
    gemm(x, W, hB, N, F, 0);                                 // h = x @ W
    hipLaunchKernelGGL(attvec_kernel, dim3((N + 255) / 256), dim3(256), 0, stream,
                       hB, avec, avec + H, f1b, f2b, N);     // f1, f2
    hipLaunchKernelGGL(att_softmax_kernel, dim3(N), dim3(256), 0, stream,
                       adj, f1b, f2b, attbuf, N);            // masked softmax
    gemm(attbuf, hB, xout, N, N, 1);                         // elu(att @ h)
  };

  // ---- start GAT (N=4096, F=128) ----
  gat(x_in, adj_in, W_start, a_start, N0, F0, att_ws, hA);

  // ---- down path with top-k pooling ----
  const float* adj_cur = adj_in;
  int Nl = N0;
  for (int i = 0; i < 3; ++i) {
    gat(hA, adj_cur, W_down + (long)i * H * H, a_down + (long)i * 2 * H, Nl, H,
        att_ws, down_arr[i]);
    hipLaunchKernelGGL(pool_scores_kernel, dim3((Nl + 255) / 256), dim3(256), 0, stream,
                       down_arr[i], p_pool + (long)i * H, scoresb, Nl);
    hipLaunchKernelGGL(topk_kernel, dim3(1), dim3(1024), 0, stream,
                       scoresb, svals, sidx, Nl);
    int k = Nl / 2;
    hipLaunchKernelGGL(gather_newx_kernel, dim3((k * H + 255) / 256), dim3(256), 0, stream,
                       down_arr[i], svals, sidx, hA, k);
    hipLaunchKernelGGL(gather_adj_kernel,
                       dim3((unsigned)(((long)k * k + 255) / 256)), dim3(256), 0, stream,
                       adj_cur, sidx, adj_next_arr[i], k, Nl);
    hipLaunchKernelGGL(compose_idx_kernel, dim3((k + 255) / 256), dim3(256), 0, stream,
                       (i == 0) ? (const int*)nullptr : (const int*)comp_arr[i - 1],
                       sidx, comp_arr[i], idx_arr[i], k);
    // gft_down[i]: zero + row scatter of pooled features
    float* gd = out_gft_down + (long)i * N0 * H;
    zero(gd, (long)N0 * H);
    hipLaunchKernelGGL(scatter_rows_kernel, dim3((k * H + 255) / 256), dim3(256), 0, stream,
                       hA, comp_arr[i], gd, k, H);
    // recon[i]: zero + 2D scatter of pooled adjacency
    float* rc = out_recon + (long)i * N0 * N0;
    zero(rc, (long)N0 * N0);
    hipLaunchKernelGGL(scatter_adj_kernel,
                       dim3((unsigned)(((long)k * k + 255) / 256)), dim3(256), 0, stream,
                       adj_next_arr[i], comp_arr[i], rc, k, N0);
    // score_list[i]
    float* sc = out_scores + (long)i * N0;
    if (i == 0) {
      copy(sc, scoresb, (long)N0);
    } else {
      zero(sc, (long)N0);
      hipLaunchKernelGGL(scatter_scalar_kernel, dim3((Nl + 255) / 256), dim3(256), 0, stream,
                         scoresb, comp_arr[i - 1], sc, Nl);
    }
    adj_cur = adj_next_arr[i];
    Nl = k;  // 2048, 1024, 512
  }

  // ---- bottom GAT (N=512) ----
  gat(hA, adj3, W_bottom, a_bottom, Nl, H, att_ws, hA);

  // ---- up path: unpool + skip + GAT ----
  const float* adjs[3] = {adj_in, adj1, adj2};
  const int Ns[4] = {4096, 2048, 1024, 512};
  for (int i = 0; i < 3; ++i) {
    int u = 2 - i;
    int Nu = Ns[u];        // 1024, 2048, 4096
    int ks = Ns[u + 1];    // rows currently held in hA: 512, 1024, 2048
    copy(hC, down_arr[u], (long)Nu * H);
    hipLaunchKernelGGL(scatter_add_rows_kernel, dim3((ks * H + 255) / 256), dim3(256), 0, stream,
                       hA, idx_arr[u], hC, ks, H);
    float* attbuf = (i == 2) ? out_att : att_ws;  // final att goes straight to d_out
    gat(hC, adjs[u], W_up + (long)i * H * H, a_up + (long)i * 2 * H, Nu, H, attbuf, hA);
    if (i < 2) {
      float* gu = out_gft_up + (long)i * N0 * H;
      zero(gu, (long)N0 * H);
      hipLaunchKernelGGL(scatter_rows_kernel, dim3((Nu * H + 255) / 256), dim3(256), 0, stream,
                         hA, comp_arr[u - 1], gu, Nu, H);
    } else {
      copy(out_gft_up + (long)2 * N0 * H, hA, (long)N0 * H);
    }
  }

  // ---- output head ----
  hipLaunchKernelGGL(out_layer_kernel, dim3((N0 * 8 + 255) / 256), dim3(256), 0, stream,
                     hA, W_out, b_out, out_pred, N0);
}